// Mu_Infor_Channel_23605140259217
// MI455X (gfx1250) — compile-verified
//
#include <hip/hip_runtime.h>
#include <math.h>

typedef __attribute__((ext_vector_type(2))) float v2f;
typedef __attribute__((ext_vector_type(8))) float v8f;

#define NB     256
#define CDIM   512
#define BDIM   4
#define PDIM   4096   // 64*64
#define MSPIX  256    // 16*16
#define EPSF   1e-8f

__device__ __forceinline__ int bin_of(float x, float mn, float mx) {
  float rel = (x - mn) / (mx - mn) * 255.0f;   // matches jnp op order
  int i = (int)rel;                            // trunc toward zero (rel >= 0)
  return i < 0 ? 0 : (i > 255 ? 255 : i);
}

// ---------------------------------------------------------------------------
// K0: tables.
//  Tt[k], k in [0,512] represents count c = k-256:
//    c >= 0 : -(c/65536)*log(c/65536+eps); c < 0 : 0 (arbitrary, cancels exactly)
//  Et[k], k in [0,256]: -(k/256)*log(k/256+eps)
// ---------------------------------------------------------------------------
__global__ void k_tables(float* __restrict__ Tt, float* __restrict__ Et) {
  int t = threadIdx.x;
  if (t < 513) {
    int c = t - 256;
    float v = 0.0f;
    if (c >= 0) {
      float p = (float)c * (1.0f / 65536.0f);
      v = -p * logf(p + EPSF);
    }
    Tt[t] = v;
  }
  if (t < 257) {
    float p = (float)t * (1.0f / 256.0f);
    Et[t] = -p * logf(p + EPSF);
  }
}

// ---------------------------------------------------------------------------
// K1: channel mean via V_WMMA_F32_16X16X4_F32.
//  f_p_m[b,p] = (1/512) * sum_c f_p[b,c,p].
//  A = all-ones 16x4 (layout-independent); B tile = 4 channels x 16 pixels.
//  D columns hold per-pixel partial sums; accumulate 128 K-chunks in f32.
//  One wave (32 threads) per 16-pixel tile; EXEC all ones at WMMA.
// ---------------------------------------------------------------------------
__global__ void k_chanmean_wmma(const float* __restrict__ fp,
                                float* __restrict__ fpm) {
  int blk  = blockIdx.x;           // BDIM * (PDIM/16) blocks
  int b    = blk >> 8;             // 256 tiles per batch
  int tile = blk & 255;
  int pbase = tile * 16;
  int lane  = threadIdx.x;         // 0..31
  int col   = lane & 15;
  int khalf = (lane >> 4) << 1;    // 0 for lanes 0-15, 2 for lanes 16-31

  const float* base = fp + (size_t)b * CDIM * PDIM + pbase + col;

  v8f acc = {};
  v2f a; a.x = 1.0f; a.y = 1.0f;

  for (int c = 0; c < CDIM; c += 4) {
    v2f bm;
    bm.x = base[(size_t)(c + khalf)     * PDIM];
    bm.y = base[(size_t)(c + khalf + 1) * PDIM];
    acc = __builtin_amdgcn_wmma_f32_16x16x4_f32(
        /*neg_a=*/false, a, /*neg_b=*/false, bm,
        /*c_mod=*/(short)0, acc, /*reuse_a=*/false, /*reuse_b=*/false);
  }
  // All D rows are identical (ones * B); VGPR0 lanes 0-15 hold row M=0.
  if (lane < 16) fpm[b * PDIM + pbase + lane] = acc[0] * (1.0f / 512.0f);
}

// ---------------------------------------------------------------------------
// K2: per-batch prep. 4 blocks x 256 threads.
//  - min/max + quantize f_p_m (4096 px) -> p-histogram -> h_p[b]
//  - 4x4 pool -> f_p_a (256), min/max + quantize -> pan[], m-hist -> mh[]
//  - u[b][v] = sum_j Tt[512 - v - mh_j], v in [0,256]
// ---------------------------------------------------------------------------
__global__ void k_batch_prep(const float* __restrict__ fpm,
                             const float* __restrict__ Tt,
                             int* __restrict__ pan_g, int* __restrict__ mh_g,
                             float* __restrict__ u_g, float* __restrict__ hp_g) {
  int b = blockIdx.x;
  int t = threadIdx.x;  // 0..255
  __shared__ float redn[256], redx[256];
  __shared__ int   ph[256], mh[256];
  __shared__ float Tts[513];
  __shared__ float mn_s, mx_s, amn_s, amx_s;

  const float* src = fpm + b * PDIM;

  Tts[t] = Tt[t]; Tts[t + 256] = Tt[t + 256];
  if (t == 0) Tts[512] = Tt[512];
  ph[t] = 0; mh[t] = 0;

  // global min/max of f_p_m over 4096
  float mn = 3.4e38f, mx = -3.4e38f;
  for (int k = 0; k < 16; ++k) {
    float x = src[t + 256 * k];
    mn = fminf(mn, x); mx = fmaxf(mx, x);
  }
  redn[t] = mn; redx[t] = mx;
  __syncthreads();
  for (int s = 128; s > 0; s >>= 1) {
    if (t < s) { redn[t] = fminf(redn[t], redn[t + s]);
                 redx[t] = fmaxf(redx[t], redx[t + s]); }
    __syncthreads();
  }
  if (t == 0) { mn_s = redn[0]; mx_s = redx[0]; }
  __syncthreads();
  float gmn = mn_s, gmx = mx_s;
  __syncthreads();

  // 4x4 average pool -> one pooled value per thread
  int pi = t >> 4, pj = t & 15;
  float s16 = 0.0f;
  for (int r = 0; r < 4; ++r)
    for (int cc = 0; cc < 4; ++cc)
      s16 += src[(pi * 4 + r) * 64 + pj * 4 + cc];
  float pa = s16 * (1.0f / 16.0f);

  redn[t] = pa; redx[t] = pa;
  __syncthreads();
  for (int s = 128; s > 0; s >>= 1) {
    if (t < s) { redn[t] = fminf(redn[t], redn[t + s]);
                 redx[t] = fmaxf(redx[t], redx[t + s]); }
    __syncthreads();
  }
  if (t == 0) { amn_s = redn[0]; amx_s = redx[0]; }
  __syncthreads();

  int pabin = bin_of(pa, amn_s, amx_s);
  pan_g[b * 256 + t] = pabin;
  atomicAdd(&mh[pabin], 1);

  // histogram of quantized f_p_m
  for (int k = 0; k < 16; ++k) {
    float x = src[t + 256 * k];
    atomicAdd(&ph[bin_of(x, gmn, gmx)], 1);
  }
  __syncthreads();

  mh_g[b * 256 + t] = mh[t];

  // h_p[b] = -sum_bins p*log(p+eps), p = ph/4096
  float pk = (float)ph[t] * (1.0f / 4096.0f);
  redn[t] = -pk * logf(pk + EPSF);
  __syncthreads();
  for (int s = 128; s > 0; s >>= 1) {
    if (t < s) redn[t] += redn[t + s];
    __syncthreads();
  }
  if (t == 0) hp_g[b] = redn[0];

  // u[b][v] = sum_j Tt_ext[256 - v - mh_j]  (index 512 - v - mh_j in [0,512])
  for (int v = t; v <= 256; v += 256) {
    float s = 0.0f;
    for (int j = 0; j < 256; ++j) s += Tts[512 - v - mh[j]];
    u_g[b * 257 + v] = s;
  }
}

// ---------------------------------------------------------------------------
// K3: per-(B,C) mutual information. 2048 blocks x 256 threads.
//  mi_raw = h_p + h_ms - h_ms_p
//         = h_p + sum_t ( Et[n_t] - u[n_t] - delta_t )
//  delta_t (first-occurrence pixels only):
//         Tt[256 + base + 2J] - Tt[256 + base], base = 256 - n_i - m_j
// ---------------------------------------------------------------------------
__global__ void k_mi(const float* __restrict__ fms,
                     const int* __restrict__ pan_g, const int* __restrict__ mh_g,
                     const float* __restrict__ u_g, const float* __restrict__ hp_g,
                     const float* __restrict__ Tt, const float* __restrict__ Et,
                     float* __restrict__ mi_raw) {
  int bc = blockIdx.x;       // b*512 + c
  int b  = bc >> 9;
  int t  = threadIdx.x;      // 0..255
  __shared__ int   msn[256], nh[256], pas[256], mhs[256];
  __shared__ float us[257], Tts[513], Ets[257], red[256];
  __shared__ float mn_s, mx_s;

  Tts[t] = Tt[t]; Tts[t + 256] = Tt[t + 256];
  if (t == 0) Tts[512] = Tt[512];
  Ets[t] = Et[t]; if (t == 0) Ets[256] = Et[256];
  us[t]  = u_g[b * 257 + t]; if (t == 0) us[256] = u_g[b * 257 + 256];
  pas[t] = pan_g[b * 256 + t];
  mhs[t] = mh_g[b * 256 + t];
  nh[t]  = 0;

  float x = fms[(size_t)bc * 256 + t];

  red[t] = x; __syncthreads();
  for (int s = 128; s > 0; s >>= 1) {
    if (t < s) red[t] = fminf(red[t], red[t + s]);
    __syncthreads();
  }
  if (t == 0) mn_s = red[0];
  __syncthreads();
  red[t] = x; __syncthreads();
  for (int s = 128; s > 0; s >>= 1) {
    if (t < s) red[t] = fmaxf(red[t], red[t + s]);
    __syncthreads();
  }
  if (t == 0) mx_s = red[0];
  __syncthreads();

  int bin = bin_of(x, mn_s, mx_s);
  msn[t] = bin;
  atomicAdd(&nh[bin], 1);
  __syncthreads();

  float hms_t  = Ets[nh[t]];   // entropy contribution of bin t
  float base_t = us[nh[t]];    // joint-entropy baseline contribution of bin t

  // sparse correction: all-pairs over the 256 pixel keys (broadcast reads)
  int i = msn[t], j = pas[t];
  int J = 0, firstq = 256;
  for (int q = 0; q < 256; ++q) {
    bool m = (msn[q] == i) && (pas[q] == j);
    if (m) { J++; if (q < firstq) firstq = q; }
  }
  float delta_t = 0.0f;
  if (firstq == t) {
    int basec = 256 - nh[i] - mhs[j];            // in [-256, 256]
    delta_t = Tts[256 + basec + 2 * J] - Tts[256 + basec];
  }

  red[t] = hms_t - base_t - delta_t;
  __syncthreads();
  for (int s = 128; s > 0; s >>= 1) {
    if (t < s) red[t] += red[t + s];
    __syncthreads();
  }
  if (t == 0) mi_raw[bc] = hp_g[b] + red[0];
}

// ---------------------------------------------------------------------------
// K4: softmax over 512 channels per batch (max-subtracted, like jax.nn.softmax)
// ---------------------------------------------------------------------------
__global__ void k_softmax(const float* __restrict__ mi_raw,
                          float* __restrict__ mif) {
  int b = blockIdx.x, t = threadIdx.x;   // 512 threads
  __shared__ float red[512];
  float v = mi_raw[b * CDIM + t];
  red[t] = v; __syncthreads();
  for (int s = 256; s > 0; s >>= 1) {
    if (t < s) red[t] = fmaxf(red[t], red[t + s]);
    __syncthreads();
  }
  float mx = red[0];
  __syncthreads();
  float e = expf(v - mx);
  red[t] = e; __syncthreads();
  for (int s = 256; s > 0; s >>= 1) {
    if (t < s) red[t] += red[t + s];
    __syncthreads();
  }
  mif[b * CDIM + t] = e / red[0];
}

// ---------------------------------------------------------------------------
// K5: out = f_ms * (1 + mi)
// ---------------------------------------------------------------------------
__global__ void k_final(const float* __restrict__ fms,
                        const float* __restrict__ mif,
                        float* __restrict__ out, int n) {
  int idx = blockIdx.x * blockDim.x + threadIdx.x;
  if (idx < n) {
    float xv = fms[idx];
    out[idx] = xv + xv * mif[idx >> 8];
  }
}

extern "C" void kernel_launch(void* const* d_in, const int* in_sizes, int n_in,
                              void* d_out, int out_size, void* d_ws, size_t ws_size,
                              hipStream_t stream) {
  (void)in_sizes; (void)n_in; (void)ws_size;
  const float* f_p  = (const float*)d_in[0];   // [4,512,64,64]
  const float* f_ms = (const float*)d_in[1];   // [4,512,16,16]
  float* out = (float*)d_out;

  // workspace carve (~97 KB, all 4B-typed; written before read every call)
  float* Tt  = (float*)d_ws;            // 513  (pad 520)
  float* Et  = Tt + 520;                // 257  (pad 264)
  float* fpm = Et + 264;                // 4*4096
  int*   pan = (int*)(fpm + BDIM * PDIM);   // 4*256
  int*   mh  = pan + BDIM * 256;            // 4*256
  float* u   = (float*)(mh + BDIM * 256);   // 4*257 (pad 1040)
  float* hp  = u + 1040;                // 4 (pad 8)
  float* mir = hp + 8;                  // 2048
  float* mif = mir + BDIM * CDIM;       // 2048

  k_tables<<<1, 544, 0, stream>>>(Tt, Et);
  k_chanmean_wmma<<<BDIM * (PDIM / 16), 32, 0, stream>>>(f_p, fpm);
  k_batch_prep<<<BDIM, 256, 0, stream>>>(fpm, Tt, pan, mh, u, hp);
  k_mi<<<BDIM * CDIM, 256, 0, stream>>>(f_ms, pan, mh, u, hp, Tt, Et, mir);
  k_softmax<<<BDIM, CDIM, 0, stream>>>(mir, mif);
  int n = out_size;
  k_final<<<(n + 255) / 256, 256, 0, stream>>>(f_ms, mif, out, n);
}